// FieldBuilder_78125455114894
// MI455X (gfx1250) — compile-verified
//
#include <hip/hip_runtime.h>

// FieldBuilder: TSC (order-3) particle->mesh scatter-add.
// out[c, x, y, z] += w27(particle) * emb[particle, c], periodic mesh n=200.
//
// Bottleneck: 21.6M f32 atomic RMWs into a 256MB grid (mostly L2-resident on
// MI455X's 192MB L2). Strategy: hardware global_atomic_add_f32 (no CAS),
// async global->LDS staging of particle data (ASYNCcnt path), wide zeroing.

#define FB_BLOCK 256
#define FB_MESH_RES 0.1f

#if defined(__gfx1250__) && \
    __has_builtin(__builtin_amdgcn_global_load_async_to_lds_b32) && \
    __has_builtin(__builtin_amdgcn_global_load_async_to_lds_b128)
#define FB_USE_ASYNC_LDS 1
#else
#define FB_USE_ASYNC_LDS 0
#endif

__device__ __forceinline__ void fb_wait_async0() {
#if defined(__gfx1250__)
#if __has_builtin(__builtin_amdgcn_s_wait_asynccnt)
  __builtin_amdgcn_s_wait_asynccnt(0);
#else
  asm volatile("s_wait_asynccnt 0x0" ::: "memory");
#endif
#endif
  asm volatile("" ::: "memory");  // keep LDS reads below the wait
}

__device__ __forceinline__ void fb_atomic_add(float* p, float v) {
  // unsafeAtomicAdd lowers to the hardware global_atomic_add_f32 (no-return
  // form when result unused) instead of a CAS loop.
  unsafeAtomicAdd(p, v);
}

__global__ void __launch_bounds__(FB_BLOCK) fb_zero(float* __restrict__ out, long n) {
  long i4 = ((long)blockIdx.x * blockDim.x + threadIdx.x) * 4;
  if (i4 + 3 < n) {
    *(float4*)(out + i4) = make_float4(0.f, 0.f, 0.f, 0.f);
  } else {
    for (long j = i4; j < n; ++j) out[j] = 0.f;
  }
}

__global__ void __launch_bounds__(FB_BLOCK) fb_scatter(
    const float* __restrict__ pos,    // [N,3]
    const float* __restrict__ cellm,  // [3,3]
    const float* __restrict__ emb,    // [N,8]
    float* __restrict__ out,          // [8, n, n, n]
    int N) {
  const int tid = threadIdx.x;
  const int p   = blockIdx.x * FB_BLOCK + tid;
  const int pc  = (p < N) ? p : (N - 1);  // clamp tail lanes to a safe address

#if FB_USE_ASYNC_LDS
  __shared__ __align__(16) float s_emb[FB_BLOCK * 8];
  __shared__ __align__(16) float s_pos[FB_BLOCK * 3];
  {
    typedef int fb_v4i __attribute__((vector_size(16)));
    typedef __attribute__((address_space(1))) int*     gi_p;
    typedef __attribute__((address_space(3))) int*     li_p;
    typedef __attribute__((address_space(1))) fb_v4i*  gv4_p;
    typedef __attribute__((address_space(3))) fb_v4i*  lv4_p;

    gi_p  gp = (gi_p)(void*)const_cast<float*>(pos + (size_t)pc * 3u);
    li_p  lp = (li_p)(void*)(&s_pos[tid * 3]);
    gv4_p ge = (gv4_p)(void*)const_cast<float*>(emb + (size_t)pc * 8u);
    lv4_p le = (lv4_p)(void*)(&s_emb[tid * 8]);

    __builtin_amdgcn_global_load_async_to_lds_b32(gp + 0, lp + 0, 0, 0);
    __builtin_amdgcn_global_load_async_to_lds_b32(gp + 1, lp + 1, 0, 0);
    __builtin_amdgcn_global_load_async_to_lds_b32(gp + 2, lp + 2, 0, 0);
    __builtin_amdgcn_global_load_async_to_lds_b128(ge + 0, le + 0, 0, 0);
    __builtin_amdgcn_global_load_async_to_lds_b128(ge + 1, le + 1, 0, 0);
  }
#endif

  // Uniform mesh parameters from the cell matrix (overlaps with the async DMA).
  const float mesh_size = (cellm[0] + cellm[4] + cellm[8]) * (1.0f / 3.0f);
  const int   nm  = 2 * (int)rintf(mesh_size / (2.0f * FB_MESH_RES));
  const float inv_spacing = (float)nm / mesh_size;
  const long  n3 = (long)nm * (long)nm * (long)nm;

  float px, py, pz, e[8];
#if FB_USE_ASYNC_LDS
  fb_wait_async0();
  px = s_pos[tid * 3 + 0];
  py = s_pos[tid * 3 + 1];
  pz = s_pos[tid * 3 + 2];
#pragma unroll
  for (int c = 0; c < 8; ++c) e[c] = s_emb[tid * 8 + c];
#else
  px = pos[(size_t)pc * 3 + 0];
  py = pos[(size_t)pc * 3 + 1];
  pz = pos[(size_t)pc * 3 + 2];
#pragma unroll
  for (int c = 0; c < 8; ++c) e[c] = emb[(size_t)pc * 8 + c];
#endif

  if (p >= N) return;

  // Nearest-grid-point centering (odd order); rintf == round-half-even == jnp.round.
  const float fx = px * inv_spacing;
  const float fy = py * inv_spacing;
  const float fz = pz * inv_spacing;
  const int ix = (int)rintf(fx);
  const int iy = (int)rintf(fy);
  const int iz = (int)rintf(fz);
  const float dx = fx - (float)ix;
  const float dy = fy - (float)iy;
  const float dz = fz - (float)iz;

  // TSC weights: 0.125(1 -4d +4d^2), 0.25(3 -4d^2), 0.125(1 +4d +4d^2)
  float wx[3], wy[3], wz[3];
  {
    const float dx2 = dx * dx, dy2 = dy * dy, dz2 = dz * dz;
    wx[0] = 0.125f * (1.0f - 4.0f * dx + 4.0f * dx2);
    wx[1] = 0.25f  * (3.0f - 4.0f * dx2);
    wx[2] = 0.125f * (1.0f + 4.0f * dx + 4.0f * dx2);
    wy[0] = 0.125f * (1.0f - 4.0f * dy + 4.0f * dy2);
    wy[1] = 0.25f  * (3.0f - 4.0f * dy2);
    wy[2] = 0.125f * (1.0f + 4.0f * dy + 4.0f * dy2);
    wz[0] = 0.125f * (1.0f - 4.0f * dz + 4.0f * dz2);
    wz[1] = 0.25f  * (3.0f - 4.0f * dz2);
    wz[2] = 0.125f * (1.0f + 4.0f * dz + 4.0f * dz2);
  }

  // Wrapped neighbor indices (idx-1 .. idx+1), idx in [0, nm].
  int gx[3], gy[3], gz[3];
#pragma unroll
  for (int a = 0; a < 3; ++a) {
    int vx = ix - 1 + a; if (vx < 0) vx += nm; if (vx >= nm) vx -= nm;
    int vy = iy - 1 + a; if (vy < 0) vy += nm; if (vy >= nm) vy -= nm;
    int vz = iz - 1 + a; if (vz < 0) vz += nm; if (vz >= nm) vz -= nm;
    gx[a] = vx; gy[a] = vy; gz[a] = vz;
  }

#pragma unroll
  for (int a = 0; a < 3; ++a) {
    const int xb = gx[a] * nm;
#pragma unroll
    for (int b = 0; b < 3; ++b) {
      const float wxy = wx[a] * wy[b];
      const int rowbase = (xb + gy[b]) * nm;
#pragma unroll
      for (int cz = 0; cz < 3; ++cz) {
        const float w = wxy * wz[cz];
        const long cell = (long)(rowbase + gz[cz]);
#pragma unroll
        for (int c = 0; c < 8; ++c) {
          fb_atomic_add(out + (long)c * n3 + cell, w * e[c]);
        }
      }
    }
  }
}

extern "C" void kernel_launch(void* const* d_in, const int* in_sizes, int n_in,
                              void* d_out, int out_size, void* d_ws, size_t ws_size,
                              hipStream_t stream) {
  const float* pos  = (const float*)d_in[0];
  const float* cell = (const float*)d_in[1];
  const float* emb  = (const float*)d_in[2];
  float* out = (float*)d_out;
  const int N = in_sizes[0] / 3;

  // 1) Zero the 256MB grid (harness poisons d_out).
  {
    long n = (long)out_size;
    long nthreads = (n + 3) / 4;
    int blocks = (int)((nthreads + FB_BLOCK - 1) / FB_BLOCK);
    fb_zero<<<blocks, FB_BLOCK, 0, stream>>>(out, n);
  }
  // 2) Scatter-add deposit.
  {
    int blocks = (N + FB_BLOCK - 1) / FB_BLOCK;
    fb_scatter<<<blocks, FB_BLOCK, 0, stream>>>(pos, cell, emb, out, N);
  }
}